// SkipGramNegativeSample_31516470018493
// MI455X (gfx1250) — compile-verified
//
#include <hip/hip_runtime.h>

typedef __attribute__((ext_vector_type(2))) float v2f;
typedef __attribute__((ext_vector_type(8))) float v8f;

#define Vn 100000
#define Dn 300
#define Bn 4096
#define Cn 10
#define NNEGn 5
#define NROW 60              // Cn + Cn*NNEGn rows per batch element
#define WPB 8                // waves (batch elements) per block
#define NBLK (Bn / WPB)      // 512 blocks
#define KCHUNKS (Dn / 4)     // 75 WMMA k-steps of 4

// One wave32 per batch element. Scores via V_WMMA_F32_16X16X4_F32 with the
// ivec chunk broadcast across the 16 N-columns (all columns of D identical).
__global__ __launch_bounds__(256) void sgns_main_kernel(
    const float* __restrict__ ivw,     // [V, D]
    const float* __restrict__ ovw,     // [V, D]
    const int*   __restrict__ iword,   // [B]
    const int*   __restrict__ owords,  // [B, C]
    const int*   __restrict__ nwords,  // [B, C*NNEG]
    float* __restrict__ partial)       // [NBLK] per-block partial sums
{
    __shared__ float s_iv[WPB][304];   // staged ivec per wave (padded stride)
    __shared__ float s_red[WPB];

    const int tid  = threadIdx.x;
    const int wave = tid >> 5;
    const int lane = tid & 31;
    const int b    = blockIdx.x * WPB + wave;

    // Stage this wave's center vector into LDS (coalesced).
    const float* ivrow = ivw + (long long)iword[b] * Dn;
    for (int j = lane; j < Dn; j += 32) s_iv[wave][j] = ivrow[j];
    __syncthreads();

    const int hi = lane >> 4;   // 0: K pair {0,1}, 1: K pair {2,3}
    const int lo = lane & 15;

    float acc_loss = 0.0f;

    for (int t = 0; t < 4; ++t) {
        // Row handled by this lane's half-group: r = t*16 + lo
        const int r = t * 16 + lo;
        int word = 0;
        if (r < Cn)          word = owords[b * Cn + r];
        else if (r < NROW)   word = nwords[b * (Cn * NNEGn) + (r - Cn)];
        // padding rows (60..63) -> word 0 (the all-zero row), masked below
        const float* rowp = ovw + (long long)word * Dn;
        __builtin_prefetch(rowp, 0, 0);   // global_prefetch_b8, pull line early

        v8f acc = {};
        #pragma unroll 5
        for (int it = 0; it < KCHUNKS; ++it) {
            const int off = 4 * it + 2 * hi;
            // A: 16x4 f32 tile — lane pair (contiguous 8B global load)
            const v2f a = *(const v2f*)(rowp + off);
            // B: 4x16 f32 tile — ivec chunk broadcast (8B LDS broadcast read)
            const v2f bb = *(const v2f*)&s_iv[wave][off];
            acc = __builtin_amdgcn_wmma_f32_16x16x4_f32(
                false, a, false, bb, (short)0, acc, false, false);
        }

        // D layout: lane (lo==0 group) VGPR v holds score of row m = v + 8*hi,
        // replicated across all 16 N-columns; take column N==0 only.
        if (lo == 0) {
            #pragma unroll
            for (int v = 0; v < 8; ++v) {
                const int rr = t * 16 + v + 8 * hi;
                if (rr < NROW) {
                    float s = acc[v];
                    if (rr >= Cn) s = -s;   // negatives: logsig(-dot)
                    // numerically stable log-sigmoid
                    acc_loss += fminf(s, 0.0f) - log1pf(__expf(-fabsf(s)));
                }
            }
        }
    }

    // wave32 reduction (only lanes 0 and 16 are nonzero)
    for (int o = 16; o >= 1; o >>= 1)
        acc_loss += __shfl_xor(acc_loss, o, 32);

    if (lane == 0) s_red[wave] = acc_loss;
    __syncthreads();
    if (tid == 0) {
        float s = 0.0f;
        #pragma unroll
        for (int w = 0; w < WPB; ++w) s += s_red[w];
        partial[blockIdx.x] = s;
    }
}

// Deterministic final reduction: 512 partials -> scalar loss.
__global__ __launch_bounds__(256) void sgns_reduce_kernel(
    const float* __restrict__ partial, float* __restrict__ out)
{
    __shared__ float s[256];
    const int t = threadIdx.x;
    s[t] = partial[t] + partial[t + 256];
    __syncthreads();
    for (int o = 128; o > 0; o >>= 1) {
        if (t < o) s[t] += s[t + o];
        __syncthreads();
    }
    if (t == 0) *out = -s[0] / (float)(Bn * Cn);
}

extern "C" void kernel_launch(void* const* d_in, const int* in_sizes, int n_in,
                              void* d_out, int out_size, void* d_ws, size_t ws_size,
                              hipStream_t stream) {
    const float* ivw    = (const float*)d_in[0];
    const float* ovw    = (const float*)d_in[1];
    const int*   iword  = (const int*)d_in[2];
    const int*   owords = (const int*)d_in[3];
    const int*   nwords = (const int*)d_in[4];
    float* out = (float*)d_out;
    float* partial = (float*)d_ws;   // needs NBLK * 4 = 2 KB scratch

    sgns_main_kernel<<<NBLK, 256, 0, stream>>>(ivw, ovw, iword, owords, nwords, partial);
    sgns_reduce_kernel<<<1, 256, 0, stream>>>(partial, out);
}